// GAT_small_56873956933640
// MI455X (gfx1250) — compile-verified
//
#include <hip/hip_runtime.h>
#include <math.h>

#define N_NODES 100000
#define N_EDGES 1600000
#define N_GRAPH 2048
#define DIN 128
#define H 16
#define BN_EPS 1e-5f

typedef __attribute__((ext_vector_type(2))) float v2f;
typedef __attribute__((ext_vector_type(8))) float v8f;

__device__ __forceinline__ float eluf(float x) { return x > 0.f ? x : (expf(x) - 1.f); }

__device__ __forceinline__ float wsum32(float v) {
#pragma unroll
    for (int o = 16; o > 0; o >>= 1) v += __shfl_xor(v, o, 32);
    return v;
}
__device__ __forceinline__ float wmax32(float v) {
#pragma unroll
    for (int o = 16; o > 0; o >>= 1) v = fmaxf(v, __shfl_xor(v, o, 32));
    return v;
}

// ---------------------------------------------------------------------------
// Fused dual-projection GEMM with fp32 WMMA: Ys = X @ Wself, Yn = X @ Wneigh.
// One wave per 16-row tile of X. A is 16xK fp32, walked in K-chunks of 4 via
// V_WMMA_F32_16X16X4_F32 with accumulator chaining (SRC2 = previous D).
//   A layout (16x4 fp32): lane L<16 holds rows M=L, K=kb+0/kb+1 in v2f;
//                         lane L>=16 holds rows M=L-16, K=kb+2/kb+3.
//   B layout (4x16 fp32): mirrored (lane<16 -> K rows 0/1, lane>=16 -> 2/3),
//                         column N = lane&15.
//   C/D layout (16x16 fp32): vgpr r -> M = r + (lane>=16 ? 8 : 0), N = lane&15.
// ---------------------------------------------------------------------------
template <int K>
__global__ void proj_wmma_kernel(const float* __restrict__ X,
                                 const float* __restrict__ Wself,
                                 const float* __restrict__ Wneigh,
                                 float* __restrict__ Ys,
                                 float* __restrict__ Yn,
                                 int nrows) {
    const int lane = threadIdx.x & 31;
    const int wave = threadIdx.x >> 5;
    const int tile = blockIdx.x * (blockDim.x >> 5) + wave;
    const int row0 = tile * 16;
    if (row0 >= nrows) return;                 // uniform per wave -> EXEC all-1s inside

    const int m    = lane & 15;                // A row within tile / B column
    const int koff = (lane >> 4) << 1;         // 0 for lanes 0-15, 2 for lanes 16-31

    const int rowA = min(row0 + m, nrows - 1); // clamp (reads only; stores guarded)
    const float* __restrict__ xrow = X + (size_t)rowA * K;

    v8f accS = {};
    v8f accN = {};

#pragma unroll
    for (int kk = 0; kk < K / 4; ++kk) {
        const int kb = kk * 4 + koff;
        v2f a;  a.x  = xrow[kb];             a.y  = xrow[kb + 1];
        v2f bs; bs.x = Wself[kb * H + m];    bs.y = Wself[(kb + 1) * H + m];
        v2f bn; bn.x = Wneigh[kb * H + m];   bn.y = Wneigh[(kb + 1) * H + m];
        accS = __builtin_amdgcn_wmma_f32_16x16x4_f32(false, a, false, bs,
                                                     (short)0, accS, false, false);
        accN = __builtin_amdgcn_wmma_f32_16x16x4_f32(false, a, false, bn,
                                                     (short)0, accN, false, false);
    }

    const int col = lane & 15;
    const int mhi = (lane >> 4) << 3;          // 0 or 8
#pragma unroll
    for (int r = 0; r < 8; ++r) {
        const int row = row0 + r + mhi;
        if (row < nrows) {
            Ys[(size_t)row * H + col] = accS[r];
            Yn[(size_t)row * H + col] = accN[r];
        }
    }
}

// ---------------------------------------------------------------------------
// Utility / graph kernels
// ---------------------------------------------------------------------------
__global__ void zero_f32_kernel(float* __restrict__ p, size_t n) {
    size_t i = (size_t)blockIdx.x * blockDim.x + threadIdx.x;
    if (i < n) p[i] = 0.f;
}

__global__ void init_bounds_kernel(int* __restrict__ gs, int* __restrict__ ge) {
    int i = blockIdx.x * blockDim.x + threadIdx.x;
    if (i < N_GRAPH) { gs[i] = N_NODES; ge[i] = 0; }
}

__global__ void degree_kernel(const int* __restrict__ dst, float* __restrict__ deg) {
    int e = blockIdx.x * blockDim.x + threadIdx.x;
    if (e < N_EDGES) atomicAdd(&deg[dst[e]], 1.f);
}

// scatter-add of 16-dim projected features along edges: agg[dst] += y[src]
__global__ void edge_agg_kernel(const int* __restrict__ src, const int* __restrict__ dst,
                                const float* __restrict__ y, float* __restrict__ agg) {
    size_t t = (size_t)blockIdx.x * blockDim.x + threadIdx.x;
    if (t < (size_t)N_EDGES * H) {
        const int e = (int)(t >> 4);
        const int j = (int)(t & 15);
        atomicAdd(&agg[(size_t)dst[e] * H + j], y[(size_t)src[e] * H + j]);
    }
}

// h = ELU(ys + agg/max(deg,1) + b1), written in place over ys
__global__ void finish1_kernel(float* __restrict__ ys, const float* __restrict__ agg,
                               const float* __restrict__ deg, const float* __restrict__ b1) {
    int n = blockIdx.x * blockDim.x + threadIdx.x;
    if (n >= N_NODES) return;
    const float invd = 1.f / fmaxf(deg[n], 1.f);
#pragma unroll
    for (int j = 0; j < H; ++j) {
        float v = ys[(size_t)n * H + j] + agg[(size_t)n * H + j] * invd + b1[j];
        ys[(size_t)n * H + j] = eluf(v);
    }
}

// hf = hs + agg/max(deg,1) + b2 ; gate = BN(hf)@Wg + bg ; per-graph node bounds
__global__ void finish2_kernel(const float* __restrict__ hs, const float* __restrict__ agg,
                               const float* __restrict__ deg, const float* __restrict__ b2,
                               float* __restrict__ hf,
                               const float* __restrict__ bng_g, const float* __restrict__ bng_b,
                               const float* __restrict__ bng_m, const float* __restrict__ bng_v,
                               const float* __restrict__ Wg, const float* __restrict__ bg,
                               const int* __restrict__ gid, float* __restrict__ gate,
                               int* __restrict__ gs, int* __restrict__ ge) {
    int n = blockIdx.x * blockDim.x + threadIdx.x;
    if (n >= N_NODES) return;
    const float invd = 1.f / fmaxf(deg[n], 1.f);
    float gacc = bg[0];
#pragma unroll
    for (int j = 0; j < H; ++j) {
        float v = hs[(size_t)n * H + j] + agg[(size_t)n * H + j] * invd + b2[j];
        hf[(size_t)n * H + j] = v;
        float bn = (v - bng_m[j]) * rsqrtf(bng_v[j] + BN_EPS) * bng_g[j] + bng_b[j];
        gacc += bn * Wg[j];
    }
    gate[n] = gacc;
    const int g = gid[n];
    atomicMin(&gs[g], n);
    atomicMax(&ge[g], n + 1);
}

// one wave32 per graph: segment softmax attention pool + max pool, z = ELU(concat)
__global__ void pool_kernel(const float* __restrict__ hf, const float* __restrict__ gate,
                            const int* __restrict__ gs, const int* __restrict__ ge,
                            float* __restrict__ z) {
    const int g = blockIdx.x;
    const int lane = threadIdx.x;
    const int s = gs[g];
    const int e = ge[g];

    float gm = -3.4e38f;
    for (int n = s + lane; n < e; n += 32) gm = fmaxf(gm, gate[n]);
    gm = wmax32(gm);

    float sa = 0.f;
    float acc[H], mx[H];
#pragma unroll
    for (int j = 0; j < H; ++j) { acc[j] = 0.f; mx[j] = -3.4e38f; }

    for (int n = s + lane; n < e; n += 32) {
        const float a = expf(gate[n] - gm);
        sa += a;
        const float* hr = hf + (size_t)n * H;
#pragma unroll
        for (int j = 0; j < H; ++j) {
            const float hv = hr[j];
            acc[j] += a * hv;
            mx[j] = fmaxf(mx[j], hv);
        }
    }
    sa = wsum32(sa);
#pragma unroll
    for (int j = 0; j < H; ++j) { acc[j] = wsum32(acc[j]); mx[j] = wmax32(mx[j]); }

    if (lane == 0) {
        const bool empty = (s >= e);
        const float inv = empty ? 0.f : 1.f / sa;
#pragma unroll
        for (int j = 0; j < H; ++j) {
            z[(size_t)g * 32 + j]      = eluf(empty ? 0.f : acc[j] * inv);
            z[(size_t)g * 32 + 16 + j] = eluf(empty ? 0.f : mx[j]);
        }
    }
}

// one wave32 per graph: BN(32) -> Linear(32,32)+ReLU -> Linear(32,32)+ReLU -> Linear(32,1)
__global__ void mlp_kernel(const float* __restrict__ z,
                           const float* __restrict__ bnf_g, const float* __restrict__ bnf_b,
                           const float* __restrict__ bnf_m, const float* __restrict__ bnf_v,
                           const float* __restrict__ Wf1, const float* __restrict__ bf1,
                           const float* __restrict__ Wf2, const float* __restrict__ bf2,
                           const float* __restrict__ Wf3, const float* __restrict__ bf3,
                           float* __restrict__ out) {
    __shared__ float sh[32];
    const int g = blockIdx.x;
    const int j = threadIdx.x;  // 32 lanes == 32 features/hidden units

    float zb = (z[(size_t)g * 32 + j] - bnf_m[j]) * rsqrtf(bnf_v[j] + BN_EPS) * bnf_g[j] + bnf_b[j];
    sh[j] = zb;
    __syncthreads();

    float s1 = bf1[j];
#pragma unroll
    for (int i = 0; i < 32; ++i) s1 += sh[i] * Wf1[i * 32 + j];
    s1 = fmaxf(s1, 0.f);
    __syncthreads();
    sh[j] = s1;
    __syncthreads();

    float s2 = bf2[j];
#pragma unroll
    for (int i = 0; i < 32; ++i) s2 += sh[i] * Wf2[i * 32 + j];
    s2 = fmaxf(s2, 0.f);

    float p = wsum32(s2 * Wf3[j]);
    if (j == 0) out[g] = p + bf3[0];
}

// ---------------------------------------------------------------------------
extern "C" void kernel_launch(void* const* d_in, const int* in_sizes, int n_in,
                              void* d_out, int out_size, void* d_ws, size_t ws_size,
                              hipStream_t stream) {
    (void)in_sizes; (void)n_in; (void)out_size; (void)ws_size;

    const float* x    = (const float*)d_in[0];
    const int*   src  = (const int*)d_in[1];
    const int*   dst  = (const int*)d_in[2];
    const int*   gid  = (const int*)d_in[3];
    const float* W1s  = (const float*)d_in[4];
    const float* W1n  = (const float*)d_in[5];
    const float* b1   = (const float*)d_in[6];
    const float* W2s  = (const float*)d_in[7];
    const float* W2n  = (const float*)d_in[8];
    const float* b2   = (const float*)d_in[9];
    const float* bng_g = (const float*)d_in[10];
    const float* bng_b = (const float*)d_in[11];
    const float* bng_m = (const float*)d_in[12];
    const float* bng_v = (const float*)d_in[13];
    const float* Wg   = (const float*)d_in[14];
    const float* bg   = (const float*)d_in[15];
    const float* bnf_g = (const float*)d_in[16];
    const float* bnf_b = (const float*)d_in[17];
    const float* bnf_m = (const float*)d_in[18];
    const float* bnf_v = (const float*)d_in[19];
    const float* Wf1  = (const float*)d_in[20];
    const float* bf1  = (const float*)d_in[21];
    const float* Wf2  = (const float*)d_in[22];
    const float* bf2  = (const float*)d_in[23];
    const float* Wf3  = (const float*)d_in[24];
    const float* bf3  = (const float*)d_in[25];
    float* out = (float*)d_out;

    // workspace layout (floats)
    float* ws   = (float*)d_ws;
    float* ysb  = ws;                                 // N*H : self proj -> h (layer1)
    float* ynb  = ysb + (size_t)N_NODES * H;          // N*H : neigh proj -> hs (layer2)
    float* hnb  = ynb + (size_t)N_NODES * H;          // N*H : layer2 neigh proj -> hf
    float* agg  = hnb + (size_t)N_NODES * H;          // N*H : scatter accumulator
    float* deg  = agg + (size_t)N_NODES * H;          // N   : degree
    float* gate = deg + N_NODES;                      // N
    float* zbuf = gate + N_NODES;                     // G*32
    int*   gs   = (int*)(zbuf + (size_t)N_GRAPH * 32);
    int*   ge   = gs + N_GRAPH;

    const int tiles = (N_NODES + 15) / 16;
    const int WAVES = 4;                              // 128-thread blocks, 4 waves/WGP-issue
    const dim3 projGrid((tiles + WAVES - 1) / WAVES);
    const dim3 projBlk(WAVES * 32);

    const size_t zcount = (size_t)N_NODES * H + N_NODES;     // agg + deg contiguous
    zero_f32_kernel<<<(unsigned)((zcount + 255) / 256), 256, 0, stream>>>(agg, zcount);
    init_bounds_kernel<<<(N_GRAPH + 255) / 256, 256, 0, stream>>>(gs, ge);

    // ---- layer 1: project first (128 -> 16), then aggregate 16-dim over edges
    proj_wmma_kernel<DIN><<<projGrid, projBlk, 0, stream>>>(x, W1s, W1n, ysb, ynb, N_NODES);
    degree_kernel<<<(N_EDGES + 255) / 256, 256, 0, stream>>>(dst, deg);
    {
        const size_t tot = (size_t)N_EDGES * H;
        edge_agg_kernel<<<(unsigned)((tot + 255) / 256), 256, 0, stream>>>(src, dst, ynb, agg);
    }
    finish1_kernel<<<(N_NODES + 255) / 256, 256, 0, stream>>>(ysb, agg, deg, b1);

    // ---- layer 2
    zero_f32_kernel<<<(unsigned)(((size_t)N_NODES * H + 255) / 256), 256, 0, stream>>>(
        agg, (size_t)N_NODES * H);
    proj_wmma_kernel<H><<<projGrid, projBlk, 0, stream>>>(ysb, W2s, W2n, ynb, hnb, N_NODES);
    {
        const size_t tot = (size_t)N_EDGES * H;
        edge_agg_kernel<<<(unsigned)((tot + 255) / 256), 256, 0, stream>>>(src, dst, hnb, agg);
    }
    finish2_kernel<<<(N_NODES + 255) / 256, 256, 0, stream>>>(
        ynb, agg, deg, b2, hnb, bng_g, bng_b, bng_m, bng_v, Wg, bg, gid, gate, gs, ge);

    // ---- pooling + readout
    pool_kernel<<<N_GRAPH, 32, 0, stream>>>(hnb, gate, gs, ge, zbuf);
    mlp_kernel<<<N_GRAPH, 32, 0, stream>>>(zbuf, bnf_g, bnf_b, bnf_m, bnf_v,
                                           Wf1, bf1, Wf2, bf2, Wf3, bf3, out);
}